// realworld_loss_34368328302911
// MI455X (gfx1250) — compile-verified
//
#include <hip/hip_runtime.h>
#include <math.h>

typedef __attribute__((ext_vector_type(2))) float v2f;
typedef __attribute__((ext_vector_type(8))) float v8f;

// ws layout (floats):
//   [0 .. B*12)            : per-batch M' = [RT_R - I | t_RT], row-major 3x4
//   [base+0]               : t_loss sum
//   [base+1]               : r_loss sum
//   [base+2]               : qn_loss sum
//   [base+3 .. base+3+B)   : sum_err per batch
//   [base+3+B .. base+3+2B): sum_mask per batch
// where base = B*12.

__device__ __forceinline__ float smooth_l1_f(float x, float y) {
  float d = fabsf(x - y);
  return (d < 1.0f) ? 0.5f * d * d : (d - 0.5f);
}

__device__ void mat2quat_dev(const float* R, float* q) {
  float R00 = R[0], R01 = R[1], R02 = R[2];
  float R10 = R[3], R11 = R[4], R12 = R[5];
  float R20 = R[6], R21 = R[7], R22 = R[8];
  float tr = R00 + R11 + R22;
  float t0 = 1.0f + tr;
  float t1 = 1.0f + R00 - R11 - R22;
  float t2 = 1.0f - R00 + R11 - R22;
  float t3 = 1.0f - R00 - R11 + R22;
  int sel;
  if (tr > 0.0f) {
    sel = 0;
  } else {
    // first-occurrence argmax over (R00, R11, R22), matching jnp.argmax
    int idx = 0;
    float m = R00;
    if (R11 > m) { m = R11; idx = 1; }
    if (R22 > m) { m = R22; idx = 2; }
    sel = 1 + idx;
  }
  float t, qq0, qq1, qq2, qq3;
  switch (sel) {
    case 0:  t = t0; qq0 = t0;        qq1 = R21 - R12; qq2 = R02 - R20; qq3 = R10 - R01; break;
    case 1:  t = t1; qq0 = R21 - R12; qq1 = t1;        qq2 = R01 + R10; qq3 = R02 + R20; break;
    case 2:  t = t2; qq0 = R02 - R20; qq1 = R01 + R10; qq2 = t2;        qq3 = R12 + R21; break;
    default: t = t3; qq0 = R10 - R01; qq1 = R02 + R20; qq2 = R12 + R21; qq3 = t3;        break;
  }
  float s = 0.5f / sqrtf(fmaxf(t, 1e-12f));
  q[0] = qq0 * s; q[1] = qq1 * s; q[2] = qq2 * s; q[3] = qq3 * s;
}

__device__ float quat_dist_dev(const float* q, const float* r) {
  // d = q * conj(r); return 2*atan2(|d_vec|, |d_w|)
  float rw = r[0], rx = -r[1], ry = -r[2], rz = -r[3];
  float qw = q[0], qx = q[1], qy = q[2], qz = q[3];
  float dw = qw * rw - (qx * rx + qy * ry + qz * rz);
  float dx = qw * rx + rw * qx + (qy * rz - qz * ry);
  float dy = qw * ry + rw * qy + (qz * rx - qx * rz);
  float dz = qw * rz + rw * qz + (qx * ry - qy * rx);
  return 2.0f * atan2f(sqrtf(dx * dx + dy * dy + dz * dz), fabsf(dw));
}

__global__ void pose_prep_kernel(const float* __restrict__ pred_t,
                                 const float* __restrict__ pred_q,
                                 const float* __restrict__ gt,
                                 const float* __restrict__ initT,
                                 float* __restrict__ ws, int B) {
  const int tid = threadIdx.x;
  const int base = B * 12;
  const int naccum = 3 + 2 * B;
  for (int i = tid; i < naccum; i += blockDim.x) ws[base + i] = 0.0f;
  __syncthreads();
  if (tid >= B) return;
  const int b = tid;

  // --- pred quaternion: qn loss + normalized rotation matrix ---
  const float* qp = pred_q + 4 * b;
  float qw = qp[0], qx = qp[1], qy = qp[2], qz = qp[3];
  float ns = qw * qw + qx * qx + qy * qy + qz * qz;
  float qn_c = (ns - 1.0f) * (ns - 1.0f);
  float inv = 1.0f / sqrtf(ns);
  qw *= inv; qx *= inv; qy *= inv; qz *= inv;
  float R[9];
  R[0] = 1.0f - 2.0f * (qy * qy + qz * qz); R[1] = 2.0f * (qx * qy - qw * qz); R[2] = 2.0f * (qx * qz + qw * qy);
  R[3] = 2.0f * (qx * qy + qw * qz); R[4] = 1.0f - 2.0f * (qx * qx + qz * qz); R[5] = 2.0f * (qy * qz - qw * qx);
  R[6] = 2.0f * (qx * qz - qw * qy); R[7] = 2.0f * (qy * qz + qw * qx); R[8] = 1.0f - 2.0f * (qx * qx + qy * qy);
  float tp[3] = {pred_t[3 * b], pred_t[3 * b + 1], pred_t[3 * b + 2]};

  // --- T_gt_exp = inv(T_pred) @ init : pR = R^T Ri, pt = R^T (ti - tp) ---
  const float* Ti = initT + 16 * b;
  float Ri[9] = {Ti[0], Ti[1], Ti[2], Ti[4], Ti[5], Ti[6], Ti[8], Ti[9], Ti[10]};
  float ti[3] = {Ti[3], Ti[7], Ti[11]};
  float pR[9], pt[3];
#pragma unroll
  for (int i = 0; i < 3; ++i) {
#pragma unroll
    for (int j = 0; j < 3; ++j) {
      float s = 0.0f;
#pragma unroll
      for (int k = 0; k < 3; ++k) s += R[3 * k + i] * Ri[3 * k + j];
      pR[3 * i + j] = s;
    }
    float s = 0.0f;
#pragma unroll
    for (int k = 0; k < 3; ++k) s += R[3 * k + i] * (ti[k] - tp[k]);
    pt[i] = s;
  }

  const float* Tg = gt + 16 * b;
  float gR[9] = {Tg[0], Tg[1], Tg[2], Tg[4], Tg[5], Tg[6], Tg[8], Tg[9], Tg[10]};
  float gtt[3] = {Tg[3], Tg[7], Tg[11]};

  // --- translation loss ---
  float tl = smooth_l1_f(pt[0], gtt[0]) + smooth_l1_f(pt[1], gtt[1]) + smooth_l1_f(pt[2], gtt[2]);

  // --- RT = inv(gt) @ [pR|pt] ; store M' = [RT_R - I | RT_t] ---
  float* M = ws + 12 * b;
#pragma unroll
  for (int i = 0; i < 3; ++i) {
#pragma unroll
    for (int j = 0; j < 3; ++j) {
      float s = 0.0f;
#pragma unroll
      for (int k = 0; k < 3; ++k) s += gR[3 * k + i] * pR[3 * k + j];
      M[4 * i + j] = s - ((i == j) ? 1.0f : 0.0f);
    }
    float s = 0.0f;
#pragma unroll
    for (int k = 0; k < 3; ++k) s += gR[3 * k + i] * (pt[k] - gtt[k]);
    M[4 * i + 3] = s;
  }

  // --- rotation loss ---
  float q_pred[4], q_gt[4];
  mat2quat_dev(pR, q_pred);
  mat2quat_dev(gR, q_gt);
  float rl = quat_dist_dev(q_pred, q_gt);

  atomicAdd(&ws[base + 0], tl);
  atomicAdd(&ws[base + 1], rl);
  atomicAdd(&ws[base + 2], qn_c);
}

// Each wave owns one batch; tiles of 16 points via V_WMMA_F32_16X16X4_F32.
//   A (16x4): rows 0..2 = [delta | t] of this batch, rest 0
//   B (4x16): column n = [x, y, z, 1] of point n
//   D (16x16): lane n (0..15), VGPRs 0..2 = (dx, dy, dz) of point n
// Loop body is branch-free: every lane does one 8-byte point load + one mask
// load per tile; role (x,y) vs (z,1) is resolved with selects, not branches.
__global__ void pc_err_wmma_kernel(const float* __restrict__ pcs,
                                   const int* __restrict__ mask,
                                   float* __restrict__ ws,
                                   int B, int N, int stride) {
  const int lane = threadIdx.x & 31;
  const int w = blockIdx.x * (blockDim.x >> 5) + (threadIdx.x >> 5);
  const int b = w % B;
  const int i0 = w / B;                    // wave index within batch, [0, stride)
  const int numTiles = (N + 15) >> 4;
  const int numPairs = (numTiles + 1) >> 1;
  const int base = B * 12;

  // A fragment: lane<16 -> row=lane, K={0,1}; lane>=16 -> row=lane-16, K={2,3}
  const float* M = ws + 12 * b;
  const int row = (lane < 16) ? lane : (lane - 16);
  const int c0 = (lane < 16) ? 0 : 2;
  const bool rvalid = row < 3;
  v2f afrag;
  afrag.x = rvalid ? M[4 * row + c0] : 0.0f;
  afrag.y = rvalid ? M[4 * row + c0 + 1] : 0.0f;

  const float* P0 = pcs + (size_t)b * (size_t)N * 3;
  const int* Mk = mask + (size_t)b * (size_t)N;
  const bool lo = (lane < 16);
  const int l15 = lane & 15;
  const int selOff = lo ? 0 : 1;          // lo reads (x,y) at 3p; hi reads (y,z) at 3p+1

  float acc = 0.0f, macc = 0.0f;
  for (int tp = i0; tp < numPairs; tp += stride) {
    const int tbase = tp << 1;

    // prefetch the next pair this wave will touch
    const int tn = tbase + (stride << 1);
    if (tn < numTiles)
      __builtin_prefetch(P0 + (size_t)((tn << 4) + l15) * 3, 0, 1);

#pragma unroll
    for (int u = 0; u < 2; ++u) {
      const int t = tbase + u;
      const int m0 = t << 4;
      const int p = m0 + l15;
      const int pc = (p < N) ? p : (N - 1);   // clamp (tail / overshoot tiles)

      // uniform 8-byte load; hi lanes' max address is 3*(N-1)+2 = 3N-1
      const size_t off = (size_t)pc * 3 + selOff;
      const float e0 = P0[off];
      const float e1 = P0[off + 1];
      v2f bfrag;
      bfrag.x = lo ? e0 : e1;               // x (lo) or z (hi)
      bfrag.y = lo ? e1 : 1.0f;             // y (lo) or homogeneous 1 (hi)

      const int mraw = Mk[pc];              // uniform mask load (hi lanes hit same lines)
      const float mf = (lo && p < N) ? (float)mraw : 0.0f;

      v8f cfrag = {};
      cfrag = __builtin_amdgcn_wmma_f32_16x16x4_f32(
          /*neg_a=*/false, afrag, /*neg_b=*/false, bfrag,
          /*c_mod=*/(short)0, cfrag, /*reuse_a=*/false, /*reuse_b=*/false);

      const float dx = cfrag[0], dy = cfrag[1], dz = cfrag[2];
      const float err = sqrtf(dx * dx + dy * dy + dz * dz);
      acc += err * mf;
      macc += mf;
    }
  }

  // wave32 reduction
  for (int off = 16; off > 0; off >>= 1) {
    acc += __shfl_xor(acc, off, 32);
    macc += __shfl_xor(macc, off, 32);
  }
  if (lane == 0) {
    atomicAdd(&ws[base + 3 + b], acc);
    atomicAdd(&ws[base + 3 + B + b], macc);
  }
}

__global__ void pose_final_kernel(const float* __restrict__ ws,
                                  float* __restrict__ out, int B) {
  const int base = B * 12;
  const float invB = 1.0f / (float)B;
  float t_loss = ws[base + 0] * invB;
  float r_loss = ws[base + 1] * invB;
  float qn_loss = ws[base + 2] * invB;
  float pc_loss = 0.0f;
  for (int b = 0; b < B; ++b) pc_loss += ws[base + 3 + b] / ws[base + 3 + B + b];
  pc_loss *= invB;
  out[0] = 1.0f * t_loss + 0.5f * r_loss + 0.5f * pc_loss + 0.5f * qn_loss;
}

extern "C" void kernel_launch(void* const* d_in, const int* in_sizes, int n_in,
                              void* d_out, int out_size, void* d_ws, size_t ws_size,
                              hipStream_t stream) {
  const float* pred_t = (const float*)d_in[0];
  const float* pred_q = (const float*)d_in[1];
  const float* pcs    = (const float*)d_in[2];
  const float* gt     = (const float*)d_in[3];
  const float* initT  = (const float*)d_in[4];
  const int*   maskp  = (const int*)d_in[5];
  float* out = (float*)d_out;
  float* ws  = (float*)d_ws;

  const int B = in_sizes[1] / 4;        // 8
  const int N = in_sizes[5] / B;        // 1,000,000

  pose_prep_kernel<<<1, 32, 0, stream>>>(pred_t, pred_q, gt, initT, ws, B);

  const int wavesPerBatch = 512;        // totalWaves = B * 512, exactly divisible
  const int blocks = (B * wavesPerBatch) / 8;  // 8 waves per 256-thread block
  pc_err_wmma_kernel<<<blocks, 256, 0, stream>>>(pcs, maskp, ws, B, N, wavesPerBatch);

  pose_final_kernel<<<1, 1, 0, stream>>>(ws, out, B);
}